// CAB_37022618091933
// MI455X (gfx1250) — compile-verified
//
#include <hip/hip_runtime.h>
#include <math.h>

// ---------------- problem constants ----------------
#define NB     4
#define DIM    64
#define HEADS  4
#define CH     (DIM / HEADS)      // 16
#define HH     256
#define WW     256
#define HW     (HH * WW)          // 65536
#define NPIX   (NB * HW)          // 262144
#define MAXOFF 3.0f

typedef __attribute__((ext_vector_type(2))) float        v2f;
typedef __attribute__((ext_vector_type(8))) float        v8f;
typedef __attribute__((ext_vector_type(4))) unsigned int v4u;
typedef __attribute__((ext_vector_type(8))) int          v8i;
typedef __attribute__((ext_vector_type(4))) int          v4i;

#if __has_builtin(__builtin_amdgcn_tensor_load_to_lds) && \
    __has_builtin(__builtin_amdgcn_s_wait_tensorcnt)
#define USE_TDM 1
#else
#define USE_TDM 0
#endif

// =====================================================================
// Depthwise 3x3 conv (SAME zero padding), optional bias, optional relu
// =====================================================================
__global__ void dwconv3_kernel(const float* __restrict__ in,
                               const float* __restrict__ w,
                               const float* __restrict__ bias,
                               float* __restrict__ out, int do_relu)
{
    int idx = blockIdx.x * blockDim.x + threadIdx.x;     // over NB*DIM*HW
    if (idx >= NB * DIM * HW) return;
    int x = idx % WW;
    int t = idx / WW;
    int y = t % HH;  t /= HH;
    int c = t % DIM;
    const float* wp = w + c * 9;
    const float* ip = in + (idx - (y * WW + x));         // base of (b,c) plane
    float acc = bias ? bias[c] : 0.0f;
#pragma unroll
    for (int ky = 0; ky < 3; ++ky) {
        int yy = y + ky - 1;
        if (yy < 0 || yy >= HH) continue;
#pragma unroll
        for (int kx = 0; kx < 3; ++kx) {
            int xx = x + kx - 1;
            if (xx < 0 || xx >= WW) continue;
            acc += wp[ky * 3 + kx] * ip[yy * WW + xx];
        }
    }
    if (do_relu) acc = fmaxf(acc, 0.0f);
    out[idx] = acc;
}

// =====================================================================
// 1x1 conv as GEMM via V_WMMA_F32_16X16X4_F32.
// One wave computes ALL COUT/16 output tiles for a 16-pixel column, so the
// activation column is read from memory exactly once (TDM -> LDS), then fed
// to the WMMAs from LDS while tiny weight tiles stream through the cache.
//
// A (16x4, weights): low lanes K=0..1, high lanes K=2..3 (ISA 7.12.2)
// B (4x16, activations): VGPR0 = K{0 lo / 2 hi}, VGPR1 = K{1 lo / 3 hi}
// C/D (16x16): VGPR r -> M=r (lo lanes), M=r+8 (hi lanes), N = lane%16
// grid.x = NPIX/16 pixel tiles, block = 32 (one wave).
// =====================================================================
template <int CIN, int COUT>
__global__ void conv1x1_wmma_kernel(const float* __restrict__ in,
                                    const float* __restrict__ w,
                                    const float* __restrict__ bias,
                                    float* __restrict__ out)
{
    constexpr int MT = COUT / 16;
    __shared__ float lds_b[CIN * 16];

    int lane = threadIdx.x;
    int lm = lane & 15, hi = lane >> 4;
    int P0  = blockIdx.x * 16;         // global pixel tile start
    int b   = P0 / HW;
    int pin = P0 % HW;                 // tiles never cross batch (HW%16==0)
    const float* ip = in + b * CIN * HW + pin;

#if USE_TDM
    {
        // ---- Tensor Data Mover: 2D tile (16 pixels x CIN rows) -> LDS ----
        unsigned           lds_off = (unsigned)(unsigned long long)(size_t)&lds_b[0];
        unsigned long long ga      = (unsigned long long)(size_t)ip;

        v4u g0;
        g0[0] = 1u;                               // count=1, is_restore=0
        g0[1] = lds_off;                          // lds_addr
        g0[2] = (unsigned)ga;                     // global_addr[31:0]
        g0[3] = (unsigned)((ga >> 32) & 0x01FFFFFFu) | (2u << 30); // addr[56:32] | type=2

        v8i g1;
        g1[0] = (int)(2u << 16);                  // data_size=2 (4 bytes), no mask/pad
        g1[1] = (int)(((unsigned)HW & 0xFFFFu) << 16);             // tensor_dim0 lo16
        g1[2] = (int)(((unsigned)CIN << 16) | ((unsigned)HW >> 16)); // dim0 hi16 | dim1 lo16
        g1[3] = (int)(16u << 16);                 // dim1 hi16 (0) | tile_dim0=16
        g1[4] = (int)CIN;                         // tile_dim1=CIN, tile_dim2=0
        g1[5] = (int)HW;                          // tensor_dim0_stride lo32
        g1[6] = 0;                                // stride0 hi16 | stride1 lo16
        g1[7] = (int)(((unsigned long long)HW * CIN) >> 16); // stride1 bits47:16

        v4i g2 = {0, 0, 0, 0};
        v4i g3 = {0, 0, 0, 0};
        v8i g4 = {0, 0, 0, 0, 0, 0, 0, 0};        // clang-23 form: extra group
        __builtin_amdgcn_tensor_load_to_lds(g0, g1, g2, g3, g4, 0);
        __builtin_amdgcn_s_wait_tensorcnt(0);
    }
#else
    for (int i = lane; i < CIN * 16; i += 32)
        lds_b[i] = ip[(i >> 4) * HW + (i & 15)];
#endif
    __syncthreads();

    v8f acc[MT] = {};
    for (int k = 0; k < CIN; k += 4) {
        v2f bb;
        bb.x = lds_b[(k + 2 * hi + 0) * 16 + lm];
        bb.y = lds_b[(k + 2 * hi + 1) * 16 + lm];
#pragma unroll
        for (int mt = 0; mt < MT; ++mt) {
            v2f a;
            a.x = w[(mt * 16 + lm) * CIN + k + 2 * hi + 0];
            a.y = w[(mt * 16 + lm) * CIN + k + 2 * hi + 1];
            acc[mt] = __builtin_amdgcn_wmma_f32_16x16x4_f32(
                          false, a, false, bb, (short)0, acc[mt], false, false);
        }
    }
#pragma unroll
    for (int mt = 0; mt < MT; ++mt) {
#pragma unroll
        for (int r = 0; r < 8; ++r) {
            int m = mt * 16 + r + 8 * hi;
            float v = acc[mt][r] + (bias ? bias[m] : 0.0f);
            out[(b * COUT + m) * HW + pin + lm] = v;
        }
    }
}

// =====================================================================
// Offset head: 1x1 conv 64->8, tanh*3, mean over heads, build sample grid
// =====================================================================
__global__ void grid_kernel(const float* __restrict__ tmp2,
                            const float* __restrict__ offp_w,
                            const float* __restrict__ offp_b,
                            float* __restrict__ grid)
{
    int idx = blockIdx.x * blockDim.x + threadIdx.x;     // over NPIX
    if (idx >= NPIX) return;
    int b = idx / HW, pin = idx % HW;
    int y = pin / WW, x = pin % WW;
    const float* ip = tmp2 + b * DIM * HW + pin;

    float acc[2 * HEADS];
#pragma unroll
    for (int j = 0; j < 2 * HEADS; ++j) acc[j] = offp_b[j];
    for (int c = 0; c < DIM; ++c) {
        float v = ip[c * HW];
#pragma unroll
        for (int j = 0; j < 2 * HEADS; ++j) acc[j] += offp_w[j * DIM + c] * v;
    }
    float mx = 0.0f, my = 0.0f;
#pragma unroll
    for (int h = 0; h < HEADS; ++h) {
        mx += tanhf(acc[2 * h + 0]) * MAXOFF;
        my += tanhf(acc[2 * h + 1]) * MAXOFF;
    }
    mx *= (1.0f / HEADS);
    my *= (1.0f / HEADS);
    float gx = -1.0f + 2.0f * x / (WW - 1);
    float gy = -1.0f + 2.0f * y / (HH - 1);
    gx = fminf(fmaxf(gx + mx / (WW * 0.5f), -1.0f), 1.0f);
    gy = fminf(fmaxf(gy + my / (HH * 0.5f), -1.0f), 1.0f);
    grid[idx * 2 + 0] = gx;
    grid[idx * 2 + 1] = gy;
}

// =====================================================================
// Bilinear sampling of y at grid (border clamp), per-pixel over channels
// =====================================================================
__global__ void sample_kernel(const float* __restrict__ feat,
                              const float* __restrict__ grid,
                              float* __restrict__ out)
{
    int idx = blockIdx.x * blockDim.x + threadIdx.x;     // over NPIX
    if (idx >= NPIX) return;
    int b = idx / HW, pin = idx % HW;
    float ix = (grid[idx * 2 + 0] + 1.0f) * 0.5f * (WW - 1);
    float iy = (grid[idx * 2 + 1] + 1.0f) * 0.5f * (HH - 1);
    float x0f = floorf(ix), y0f = floorf(iy);
    float wx = ix - x0f, wy = iy - y0f;
    int x0 = min(max((int)x0f, 0), WW - 1);
    int x1 = min(x0 + 1, WW - 1);
    int y0 = min(max((int)y0f, 0), HH - 1);
    int y1 = min(y0 + 1, HH - 1);
    float w00 = (1 - wx) * (1 - wy), w01 = wx * (1 - wy);
    float w10 = (1 - wx) * wy,       w11 = wx * wy;
    const float* f = feat + b * DIM * HW;
    for (int c = 0; c < DIM; ++c) {
        const float* fc = f + c * HW;
        out[(b * DIM + c) * HW + pin] =
            fc[y0 * WW + x0] * w00 + fc[y0 * WW + x1] * w01 +
            fc[y1 * WW + x0] * w10 + fc[y1 * WW + x1] * w11;
    }
}

// =====================================================================
// Inverse L2 norms of q rows (first NB*DIM) and k rows (next NB*DIM)
// =====================================================================
__global__ void norm_kernel(const float* __restrict__ q,
                            const float* __restrict__ kv,
                            float* __restrict__ invn)
{
    int row = blockIdx.x;                                // 0..2*NB*DIM-1
    const float* src;
    if (row < NB * DIM) {
        src = q + row * HW;
    } else {
        int r = row - NB * DIM;
        int b = r / DIM, c = r % DIM;
        src = kv + (b * 2 * DIM + c) * HW;               // k half of kv
    }
    float s = 0.0f;
    for (int i = threadIdx.x; i < HW; i += blockDim.x) {
        float v = src[i];
        s += v * v;
    }
    __shared__ float red[256];
    red[threadIdx.x] = s;
    __syncthreads();
    for (int st = 128; st > 0; st >>= 1) {
        if (threadIdx.x < st) red[threadIdx.x] += red[threadIdx.x + st];
        __syncthreads();
    }
    if (threadIdx.x == 0)
        invn[row] = 1.0f / fmaxf(sqrtf(red[0]), 1e-12f);
}

__global__ void zero_kernel(float* __restrict__ p, int n)
{
    int i = blockIdx.x * blockDim.x + threadIdx.x;
    if (i < n) p[i] = 0.0f;
}

// =====================================================================
// attn_raw[b,h,c,d] = sum_n q[b,h,c,n]*k[b,h,d,n]   (K = HW, split + atomics)
// grid: (bh = 16, seg = NSEG), block = 32 (one wave)
// =====================================================================
#define NSEG 32
__global__ void attn_qk_wmma_kernel(const float* __restrict__ q,
                                    const float* __restrict__ kv,
                                    float* __restrict__ attn_raw)
{
    int lane = threadIdx.x;
    int lm = lane & 15, hi = lane >> 4;
    int bh = blockIdx.x;
    int b = bh / HEADS, h = bh % HEADS;
    const int SEGLEN = HW / NSEG;
    int k0 = blockIdx.y * SEGLEN;

    const float* qp = q  + (b * DIM + h * CH + lm) * HW;        // A row (c = lm)
    const float* kp = kv + (b * 2 * DIM + h * CH + lm) * HW;    // B col (d = lm)

    v8f acc = {};
    for (int kk = k0; kk < k0 + SEGLEN; kk += 4) {
        v2f a, bb;
        a.x  = qp[kk + 2 * hi + 0];
        a.y  = qp[kk + 2 * hi + 1];
        bb.x = kp[kk + 2 * hi + 0];
        bb.y = kp[kk + 2 * hi + 1];
        acc = __builtin_amdgcn_wmma_f32_16x16x4_f32(
                  false, a, false, bb, (short)0, acc, false, false);
    }
    float* ap = attn_raw + bh * 256;
#pragma unroll
    for (int r = 0; r < 8; ++r) {
        int c = r + 8 * hi;                 // M index (q channel)
        atomicAdd(&ap[c * 16 + lm], acc[r]);
    }
}

// =====================================================================
// scale by inv-norms & temperature, softmax over d (16-wide)
// =====================================================================
__global__ void softmax_kernel(const float* __restrict__ raw,
                               const float* __restrict__ invn,
                               const float* __restrict__ temp,
                               float* __restrict__ attn)
{
    int r = blockIdx.x * blockDim.x + threadIdx.x;       // 0..NB*DIM-1
    if (r >= NB * DIM) return;
    int b = r / DIM, ci = r % DIM;
    int h = ci / CH, c = ci % CH;
    int bh = b * HEADS + h;
    float t  = temp[h];
    float iq = invn[b * DIM + h * CH + c];
    const float* rp = raw + bh * 256 + c * 16;
    float vals[16];
    float mx = -1e30f;
#pragma unroll
    for (int d = 0; d < 16; ++d) {
        float ik = invn[NB * DIM + b * DIM + h * CH + d];
        float v = rp[d] * iq * ik * t;
        vals[d] = v;
        mx = fmaxf(mx, v);
    }
    float s = 0.0f;
#pragma unroll
    for (int d = 0; d < 16; ++d) {
        vals[d] = expf(vals[d] - mx);
        s += vals[d];
    }
    float inv = 1.0f / s;
    float* op = attn + bh * 256 + c * 16;
#pragma unroll
    for (int d = 0; d < 16; ++d) op[d] = vals[d] * inv;
}

// =====================================================================
// out[b,h,c,n] = sum_d attn[b,h,c,d] * v[b,h,d,n]    (M=16,K=16,N=HW)
// grid: (HW/16 pixel tiles, bh = 16), block = 32 (one wave)
// =====================================================================
__global__ void attn_v_wmma_kernel(const float* __restrict__ attn,
                                   const float* __restrict__ kv,
                                   float* __restrict__ out)
{
    int lane = threadIdx.x;
    int lm = lane & 15, hi = lane >> 4;
    int bh = blockIdx.y;
    int b = bh / HEADS, h = bh % HEADS;
    int pin = blockIdx.x * 16;
    const float* ap = attn + bh * 256;
    const float* vp = kv + (b * 2 * DIM + DIM + h * CH) * HW + pin;  // v half

    v8f acc = {};
#pragma unroll
    for (int k = 0; k < 16; k += 4) {
        v2f a, bb;
        a.x  = ap[lm * 16 + k + 2 * hi + 0];
        a.y  = ap[lm * 16 + k + 2 * hi + 1];
        bb.x = vp[(k + 2 * hi + 0) * HW + lm];
        bb.y = vp[(k + 2 * hi + 1) * HW + lm];
        acc = __builtin_amdgcn_wmma_f32_16x16x4_f32(
                  false, a, false, bb, (short)0, acc, false, false);
    }
#pragma unroll
    for (int r = 0; r < 8; ++r) {
        int m = h * CH + r + 8 * hi;
        out[(b * DIM + m) * HW + pin + lm] = acc[r];
    }
}

// =====================================================================
// Host-side orchestration
// =====================================================================
extern "C" void kernel_launch(void* const* d_in, const int* in_sizes, int n_in,
                              void* d_out, int out_size, void* d_ws, size_t ws_size,
                              hipStream_t stream)
{
    (void)in_sizes; (void)n_in; (void)out_size; (void)ws_size;

    const float* x      = (const float*)d_in[0];
    const float* y      = (const float*)d_in[1];
    const float* temp   = (const float*)d_in[2];
    const float* q_w    = (const float*)d_in[3];
    const float* q_b    = (const float*)d_in[4];
    const float* qdw_w  = (const float*)d_in[5];
    const float* qdw_b  = (const float*)d_in[6];
    const float* kv_w   = (const float*)d_in[7];
    const float* kv_b   = (const float*)d_in[8];
    const float* off1_w = (const float*)d_in[9];
    const float* off2_w = (const float*)d_in[10];
    const float* offp_w = (const float*)d_in[11];
    const float* offp_b = (const float*)d_in[12];
    const float* proj_w = (const float*)d_in[13];
    const float* proj_b = (const float*)d_in[14];
    float* out = (float*)d_out;

    // workspace layout (floats)
    const size_t F = (size_t)NB * DIM * HW;              // 16.78M
    float* ws        = (float*)d_ws;
    float* buf0      = ws;                               // tmp1 -> qpre -> attnout
    float* buf1      = buf0 + F;                         // tmp2 -> ydef
    float* buf2      = buf1 + F;                         // q
    float* kvbuf     = buf2 + F;                         // 2F
    float* gridbuf   = kvbuf + 2 * F;                    // NPIX*2
    float* invn      = gridbuf + (size_t)NPIX * 2;       // 2*NB*DIM
    float* attn_raw  = invn + 2 * NB * DIM;              // 16*256
    float* attn      = attn_raw + NB * HEADS * 256;      // 16*256

    const int elemBlocks = (NB * DIM * HW + 255) / 256;
    const int pixBlocks  = (NPIX + 255) / 256;

    // 1-2) offset feature path: relu(dw3(y)) -> relu(dw3(.))
    dwconv3_kernel<<<elemBlocks, 256, 0, stream>>>(y, off1_w, nullptr, buf0, 1);
    dwconv3_kernel<<<elemBlocks, 256, 0, stream>>>(buf0, off2_w, nullptr, buf1, 1);

    // 3) offsets -> sampling grid
    grid_kernel<<<pixBlocks, 256, 0, stream>>>(buf1, offp_w, offp_b, gridbuf);

    // 4-5) q path: conv1x1 (WMMA + TDM) then depthwise 3x3
    conv1x1_wmma_kernel<DIM, DIM><<<NPIX / 16, 32, 0, stream>>>(x, q_w, q_b, buf0);
    dwconv3_kernel<<<elemBlocks, 256, 0, stream>>>(buf0, qdw_w, qdw_b, buf2, 0);

    // 6) deformable sampling of y
    sample_kernel<<<pixBlocks, 256, 0, stream>>>(y, gridbuf, buf1);

    // 7) kv = conv1x1 64->128 (WMMA + TDM)
    conv1x1_wmma_kernel<DIM, 2 * DIM><<<NPIX / 16, 32, 0, stream>>>(
        buf1, kv_w, kv_b, kvbuf);

    // 8) inverse L2 norms of q and k rows
    norm_kernel<<<2 * NB * DIM, 256, 0, stream>>>(buf2, kvbuf, invn);

    // 9) attn_raw = qh . kh^T  (WMMA, K-split with atomic combine)
    zero_kernel<<<(NB * HEADS * 256 + 255) / 256, 256, 0, stream>>>(
        attn_raw, NB * HEADS * 256);
    attn_qk_wmma_kernel<<<dim3(NB * HEADS, NSEG), 32, 0, stream>>>(
        buf2, kvbuf, attn_raw);

    // 10) scale + softmax
    softmax_kernel<<<(NB * DIM + 255) / 256, 256, 0, stream>>>(
        attn_raw, invn, temp, attn);

    // 11) out = attn @ v (WMMA)
    attn_v_wmma_kernel<<<dim3(HW / 16, NB * HEADS), 32, 0, stream>>>(
        attn, kvbuf, buf0);

    // 12) final projection conv1x1 (WMMA + TDM) -> d_out
    conv1x1_wmma_kernel<DIM, DIM><<<NPIX / 16, 32, 0, stream>>>(
        buf0, proj_w, proj_b, out);
}